// MACELayer_41102837023332
// MI455X (gfx1250) — compile-verified
//
#include <hip/hip_runtime.h>

// MACE equivariant layer for MI455X (gfx1250), fp32 throughout (matches reference).
// Pipeline: zero(agg) -> edge kernel (gather + CG product + native-f32-atomic scatter,
//           L2-resident working set) -> node kernel (WMMA f32 16x16x4 for channel
//           mix + residual GEMMs, tiles staged in LDS).

#define NN   30000
#define KK   32
#define LL   9
#define EE   480000
#define AA   10
#define RB   8
#define NCORR 3
#define INV_DEG (1.0f/16.0f)
#define KL   (KK*LL)          // 288 floats per node

typedef __attribute__((ext_vector_type(2))) float v2f;
typedef __attribute__((ext_vector_type(8))) float v8f;

// Wave-local LDS ordering fence: LDS ops from one wave are in-order (DScnt),
// s_wait_dscnt 0 + memory clobber makes write->read within the wave safe.
__device__ __forceinline__ void lds_fence_wave() {
  asm volatile("s_wait_dscnt 0" ::: "memory");
}

// Native fp32 global atomic add (GLOBAL_ATOMIC_ADD_F32); avoids CAS-loop lowering.
__device__ __forceinline__ void atomic_add_f32(float* p, float v) {
  unsafeAtomicAdd(p, v);
}

// ---------------------------------------------------------------------------
__global__ void mace_zero_kernel(float4* __restrict__ p, int n4) {
  int i = blockIdx.x * blockDim.x + threadIdx.x;
  int stride = gridDim.x * blockDim.x;
  const float4 z = {0.0f, 0.0f, 0.0f, 0.0f};
  for (; i < n4; i += stride) p[i] = z;
}

// ---------------------------------------------------------------------------
// One wave (32 lanes) per edge, lane = channel k (K==32==wave32).
// msgs[e,k,p] = rw[e,k] * sum_q sH[p][q] * h_send[e,k,q],
// sH[p][q]    = sum_r C[p,q,r] * sph[e,r]   (built cooperatively in LDS).
__global__ __launch_bounds__(256)
void mace_edge_kernel(const float* __restrict__ node_features,
                      const float* __restrict__ edge_features,
                      const float* __restrict__ sph,
                      const int*   __restrict__ senders,
                      const int*   __restrict__ receivers,
                      const float* __restrict__ W_radial,
                      const float* __restrict__ Cten,
                      float*       __restrict__ agg)
{
  __shared__ float sC[729];
  __shared__ float sWr[RB*KK];
  __shared__ float sSH[8][81];

  for (int i = threadIdx.x; i < 729; i += 256)   sC[i]  = Cten[i];
  for (int i = threadIdx.x; i < RB*KK; i += 256) sWr[i] = W_radial[i];
  __syncthreads();

  const int wv   = threadIdx.x >> 5;
  const int lane = threadIdx.x & 31;
  const int e    = blockIdx.x * 8 + wv;     // E divisible by 8: all waves active
  if (e >= EE) return;

  float sphr[LL];
  #pragma unroll
  for (int r = 0; r < LL; ++r) sphr[r] = sph[e*LL + r];

  // cooperative sH build: 81 entries across 32 lanes
  for (int pq = lane; pq < 81; pq += 32) {
    float s = 0.0f;
    #pragma unroll
    for (int r = 0; r < LL; ++r) s += sC[pq*LL + r] * sphr[r];
    sSH[wv][pq] = s;
  }

  // radial weight for channel k = lane
  float rw = 0.0f;
  #pragma unroll
  for (int r = 0; r < RB; ++r) rw += edge_features[e*RB + r] * sWr[r*KK + lane];

  const int snd = senders[e];
  const int rcv = receivers[e];

  float h[LL];
  #pragma unroll
  for (int q = 0; q < LL; ++q) h[q] = node_features[(snd*KK + lane)*LL + q];

  lds_fence_wave();  // sSH visible to all lanes of this wave

  float* dst = agg + (rcv*KK + lane)*LL;
  #pragma unroll
  for (int p = 0; p < LL; ++p) {
    float s = 0.0f;
    #pragma unroll
    for (int q = 0; q < LL; ++q) s += sSH[wv][p*LL + q] * h[q];
    atomic_add_f32(&dst[p], rw * s);
  }
}

// ---------------------------------------------------------------------------
// Residual GEMM for one l-group (MC = number of lm components with this l):
// update[n,j,m] = sum_{ak} (attr[n,a]*t[n,k,m]) * W_res[a,k,j,l]   (ak = a*32+k, 320)
// B (weights) hoisted across the MC accumulators; 80 chained f32 WMMAs each.
template<int MC>
__device__ __forceinline__ void residual_group(
    const float* __restrict__ T, const float* __restrict__ At,
    const float* __restrict__ W_res, float* __restrict__ out,
    int n0, int m0, int l, int hi, int lo)
{
  const v8f vzero = {0,0,0,0,0,0,0,0};
  #pragma unroll 1
  for (int jh = 0; jh < 2; ++jh) {
    const int j = jh*16 + lo;
    v8f acc[MC];
    #pragma unroll
    for (int mi = 0; mi < MC; ++mi) acc[mi] = vzero;

    #pragma unroll 1
    for (int c = 0; c < 80; ++c) {
      const int ak0 = c*4 + hi*2;
      const int ak1 = ak0 + 1;
      const int a0 = ak0 >> 5, k0 = ak0 & 31;
      const int a1 = ak1 >> 5, k1 = ak1 & 31;
      v2f b;                                   // shared across the MC wmmas
      b.x = W_res[(ak0*KK + j)*3 + l];
      b.y = W_res[(ak1*KK + j)*3 + l];
      const float at0 = At[lo*AA + a0];
      const float at1 = At[lo*AA + a1];
      #pragma unroll
      for (int mi = 0; mi < MC; ++mi) {
        const int m = m0 + mi;
        v2f a;
        a.x = at0 * T[lo*KL + k0*LL + m];
        a.y = at1 * T[lo*KL + k1*LL + m];
        acc[mi] = __builtin_amdgcn_wmma_f32_16x16x4_f32(
            false, a, false, b, (short)0, acc[mi], false, false);
      }
    }

    #pragma unroll
    for (int mi = 0; mi < MC; ++mi) {
      const int m = m0 + mi;
      #pragma unroll
      for (int i = 0; i < 8; ++i) {
        const int node = i + 8*hi;
        out[(n0 + node)*KL + j*LL + m] = T[node*KL + j*LL + m] + acc[mi][i];
      }
    }
  }
}

// ---------------------------------------------------------------------------
// One wave per 16-node tile (2 waves / 64-thread block). N = 1875 * 16 exactly.
__global__ __launch_bounds__(64)
void mace_node_kernel(const float* __restrict__ agg,
                      const float* __restrict__ node_attr,
                      const float* __restrict__ W_lin,
                      const float* __restrict__ Cten,
                      const float* __restrict__ W_contr,
                      const float* __restrict__ W_res,
                      float*       __restrict__ out)
{
  __shared__ float sWlin[KK*KK];            // 4 KB
  __shared__ float sC[729];                 // ~2.9 KB
  __shared__ float sWc[NCORR*AA*KK];        // 3.8 KB
  __shared__ float sTile[2][16*KL];         // 36 KB (per-wave 16x32x9 tile)
  __shared__ float sAttr[2][16*AA];

  for (int i = threadIdx.x; i < KK*KK; i += 64)       sWlin[i] = W_lin[i];
  for (int i = threadIdx.x; i < 729; i += 64)         sC[i]    = Cten[i];
  for (int i = threadIdx.x; i < NCORR*AA*KK; i += 64) sWc[i]   = W_contr[i];
  __syncthreads();

  const int wv   = threadIdx.x >> 5;
  const int lane = threadIdx.x & 31;
  const int tile = blockIdx.x * 2 + wv;     // wave-uniform guard: EXEC stays full
  if (tile * 16 >= NN) return;
  const int n0 = tile * 16;

  float* T  = sTile[wv];
  float* At = sAttr[wv];

  // ---- stage agg tile (scaled by 1/avg_degree, b128 path) and attributes ----
  {
    const float4* src4 = (const float4*)(agg + n0*KL);   // 16B aligned: n0*1152 bytes
    float4*       dst4 = (float4*)T;
    for (int i = lane; i < (16*KL)/4; i += 32) {         // 1152 float4s
      float4 v = src4[i];
      v.x *= INV_DEG; v.y *= INV_DEG; v.z *= INV_DEG; v.w *= INV_DEG;
      dst4[i] = v;
    }
    for (int i = lane; i < 16*AA; i += 32) At[i] = node_attr[n0*AA + i];
  }
  lds_fence_wave();

  const int hi = lane >> 4;   // which K-pair of the WMMA A/B operand
  const int lo = lane & 15;   // row (node) / col index

  // ---- h[n,j,l] = sum_k agg[n,k,l] * W_lin[k,j]  (WMMA f32 16x16x4, in place per l) ----
  #pragma unroll 1
  for (int l = 0; l < LL; ++l) {
    v8f acc0 = {0,0,0,0,0,0,0,0};
    v8f acc1 = {0,0,0,0,0,0,0,0};
    #pragma unroll
    for (int kc = 0; kc < 8; ++kc) {
      const int k0 = kc*4 + hi*2;
      v2f a;
      a.x = T[lo*KL + k0*LL + l];
      a.y = T[lo*KL + (k0+1)*LL + l];
      v2f b0, b1;
      b0.x = sWlin[k0*KK + lo];
      b0.y = sWlin[(k0+1)*KK + lo];
      b1.x = sWlin[k0*KK + 16 + lo];
      b1.y = sWlin[(k0+1)*KK + 16 + lo];
      acc0 = __builtin_amdgcn_wmma_f32_16x16x4_f32(false, a, false, b0, (short)0, acc0, false, false);
      acc1 = __builtin_amdgcn_wmma_f32_16x16x4_f32(false, a, false, b1, (short)0, acc1, false, false);
    }
    lds_fence_wave();  // all reads of column l done before in-place overwrite
    #pragma unroll
    for (int i = 0; i < 8; ++i) {
      T[(i + 8*hi)*KL + lo*LL + l]      = acc0[i];
      T[(i + 8*hi)*KL + (16+lo)*LL + l] = acc1[i];
    }
    lds_fence_wave();
  }

  // ---- symmetric contraction (correlation 3), lane = channel k ----
  #pragma unroll 1
  for (int i = 0; i < 16; ++i) {
    float h[LL];
    #pragma unroll
    for (int p = 0; p < LL; ++p) h[p] = T[i*KL + lane*LL + p];
    float attr[AA];
    #pragma unroll
    for (int a = 0; a < AA; ++a) attr[a] = At[i*AA + a];
    float w[NCORR];
    #pragma unroll
    for (int c = 0; c < NCORR; ++c) {
      float s = 0.0f;
      #pragma unroll
      for (int a = 0; a < AA; ++a) s += attr[a] * sWc[(c*AA + a)*KK + lane];
      w[c] = s;
    }
    float t[LL];
    #pragma unroll
    for (int p = 0; p < LL; ++p) t[p] = w[0] * h[p];
    #pragma unroll 1
    for (int c = 1; c < NCORR; ++c) {
      float tn[LL];
      #pragma unroll 1
      for (int p = 0; p < LL; ++p) {
        float acc = w[c] * h[p];
        #pragma unroll 1
        for (int q = 0; q < LL; ++q) {
          const float tq = t[q];
          #pragma unroll
          for (int r = 0; r < LL; ++r)
            acc += sC[(p*LL + q)*LL + r] * (tq * h[r]);   // uniform sC addr: LDS broadcast
        }
        tn[p] = acc;
      }
      #pragma unroll
      for (int p = 0; p < LL; ++p) t[p] = tn[p];
    }
    #pragma unroll
    for (int p = 0; p < LL; ++p) T[i*KL + lane*LL + p] = t[p];   // overwrite h with t
  }
  lds_fence_wave();

  // ---- residual FCTP + output: grouped by l (W_res has only 3 unique l slices) ----
  residual_group<1>(T, At, W_res, out, n0, 0, 0, hi, lo);   // m = 0        (l=0)
  residual_group<3>(T, At, W_res, out, n0, 1, 1, hi, lo);   // m = 1..3     (l=1)
  residual_group<5>(T, At, W_res, out, n0, 4, 2, hi, lo);   // m = 4..8     (l=2)
}

// ---------------------------------------------------------------------------
extern "C" void kernel_launch(void* const* d_in, const int* in_sizes, int n_in,
                              void* d_out, int out_size, void* d_ws, size_t ws_size,
                              hipStream_t stream)
{
  const float* node_features = (const float*)d_in[0];
  const float* node_attr     = (const float*)d_in[1];
  const float* edge_features = (const float*)d_in[2];
  const float* sph           = (const float*)d_in[3];
  const int*   senders       = (const int*)d_in[4];
  const int*   receivers     = (const int*)d_in[5];
  const float* W_radial      = (const float*)d_in[6];
  const float* Cten          = (const float*)d_in[7];
  const float* W_lin         = (const float*)d_in[8];
  const float* W_contr       = (const float*)d_in[9];
  const float* W_res         = (const float*)d_in[10];
  float* out = (float*)d_out;
  float* agg = (float*)d_ws;                 // N*K*L floats = 34.56 MB scratch

  const int aggN4 = (NN*KK*LL) / 4;          // 8.64M floats, divisible by 4
  mace_zero_kernel<<<2048, 256, 0, stream>>>((float4*)agg, aggN4);
  mace_edge_kernel<<<EE/8, 256, 0, stream>>>(node_features, edge_features, sph,
                                             senders, receivers, W_radial, Cten, agg);
  mace_node_kernel<<<(NN/16 + 1)/2, 64, 0, stream>>>(agg, node_attr, W_lin, Cten,
                                                     W_contr, W_res, out);
}